// EnergyCoulomb_28003186770545
// MI455X (gfx1250) — compile-verified
//
#include <hip/hip_runtime.h>
#include <hip/hip_bf16.h>

#define NBINS 128          // padded molecule bins (n_mol = 100 <= 128)
#define KE_HALF 7.1998225f // 0.5 * 14.399645
#define CUTOFF2 100.0f     // CUTOFF^2
#define SHIFT 0.1f         // 1/CUTOFF

typedef __attribute__((ext_vector_type(2))) float v2f;
typedef __attribute__((ext_vector_type(8))) float v8f;

__device__ __forceinline__ void process_pair(const float* __restrict__ q,
                                             const int* __restrict__ idx_m,
                                             float* __restrict__ bins,
                                             int i, int j,
                                             float x, float y, float z) {
    float d2 = fmaf(x, x, fmaf(y, y, z * z));
    if (d2 <= CUTOFF2) {
        float rinv = rsqrtf(d2);                       // v_rsq_f32
        // pot = 1/d + s^2*d - 2s = rinv*(1 + s^2*d2) - 2s
        float pot = fmaf(rinv, fmaf(SHIFT * SHIFT, d2, 1.0f), -2.0f * SHIFT);
        float val = q[i] * q[j] * pot;                 // L2-resident gathers
        atomicAdd(&bins[idx_m[i]], val);               // ds_add_f32
    }
}

__global__ __launch_bounds__(256) void pair_kernel(const float* __restrict__ q,
                                                   const float* __restrict__ r_ij,
                                                   const int* __restrict__ idx_i,
                                                   const int* __restrict__ idx_j,
                                                   const int* __restrict__ idx_m,
                                                   float* __restrict__ partial,
                                                   int n_pairs) {
    __shared__ float bins[NBINS];
    for (int t = threadIdx.x; t < NBINS; t += blockDim.x) bins[t] = 0.0f;
    __syncthreads();

    const int gtid = blockIdx.x * blockDim.x + threadIdx.x;
    const int nthreads = gridDim.x * blockDim.x;
    const int n4 = n_pairs >> 2;

    // 4 pairs per iteration: 3x float4 (r_ij) + 2x int4 (indices) wide loads
    for (int c = gtid; c < n4; c += nthreads) {
        const int p0 = c * 4;
        const float4* rp = (const float4*)(r_ij + 3 * (size_t)p0);
        float4 v0 = rp[0];
        float4 v1 = rp[1];
        float4 v2 = rp[2];
        int4 ii = *(const int4*)(idx_i + p0);
        int4 jj = *(const int4*)(idx_j + p0);
        process_pair(q, idx_m, bins, ii.x, jj.x, v0.x, v0.y, v0.z);
        process_pair(q, idx_m, bins, ii.y, jj.y, v0.w, v1.x, v1.y);
        process_pair(q, idx_m, bins, ii.z, jj.z, v1.z, v1.w, v2.x);
        process_pair(q, idx_m, bins, ii.w, jj.w, v2.y, v2.z, v2.w);
    }
    // tail (n_pairs not multiple of 4): first rem global threads take one pair each
    const int rem = n_pairs - (n4 << 2);
    if (gtid < rem) {
        const int p = (n4 << 2) + gtid;
        const float* r = r_ij + 3 * (size_t)p;
        process_pair(q, idx_m, bins, idx_i[p], idx_j[p], r[0], r[1], r[2]);
    }

    __syncthreads();
    // every block fully writes its own NBINS-wide row -> no workspace memset needed
    float* row = partial + (size_t)blockIdx.x * NBINS;
    for (int t = threadIdx.x; t < NBINS; t += blockDim.x) row[t] = bins[t];
}

// Column-sum of partial[P][NBINS] on the matrix unit:
//   D(16x16) += ones(16x4) x B(4x16), accumulated over P/4 row-chunks.
// A == all-ones makes the result invariant to which K-row lands in which
// lane-half/VGPR, so loading rows {r0,r0+1} per half-wave is sum-correct.
// One wave per 16-column tile; EXEC all ones as WMMA requires.
__global__ __launch_bounds__(32) void reduce_wmma_kernel(const float* __restrict__ partial,
                                                         float* __restrict__ out,
                                                         int nchunks, int n_mol) {
    const int lane = threadIdx.x;            // 0..31, wave32
    const int col = blockIdx.x * 16 + (lane & 15);
    const int half = lane >> 4;              // 0: rows {0,1}; 1: rows {2,3}

    v2f a;
    a.x = 1.0f;
    a.y = 1.0f;
    v8f acc = {};
    for (int c = 0; c < nchunks; ++c) {
        const int r0 = c * 4 + half * 2;
        v2f b;
        b.x = partial[(size_t)r0 * NBINS + col];
        b.y = partial[(size_t)(r0 + 1) * NBINS + col];
        acc = __builtin_amdgcn_wmma_f32_16x16x4_f32(
            /*neg_a=*/false, a, /*neg_b=*/false, b,
            /*c_mod=*/(short)0, acc, /*reuse_a=*/false, /*reuse_b=*/false);
    }
    // D row M=0 lives in acc[0], lanes 0..15 (N = lane)
    if (lane < 16 && col < n_mol) out[col] = KE_HALF * acc[0];
}

extern "C" void kernel_launch(void* const* d_in, const int* in_sizes, int n_in,
                              void* d_out, int out_size, void* d_ws, size_t ws_size,
                              hipStream_t stream) {
    (void)n_in;
    const float* q     = (const float*)d_in[0];
    const float* r_ij  = (const float*)d_in[1];
    const int*   idx_i = (const int*)d_in[2];
    const int*   idx_j = (const int*)d_in[3];
    const int*   idx_m = (const int*)d_in[4];
    float* out = (float*)d_out;

    const int n_pairs = in_sizes[1] / 3;
    const int n_mol = out_size;

    // P partial rows of NBINS floats in workspace; P multiple of 4, capped at 1024
    int P = (int)(ws_size / (NBINS * sizeof(float)));
    if (P > 1024) P = 1024;
    P &= ~3;
    if (P < 4) P = 4;

    float* partial = (float*)d_ws;
    pair_kernel<<<P, 256, 0, stream>>>(q, r_ij, idx_i, idx_j, idx_m, partial, n_pairs);
    reduce_wmma_kernel<<<8, 32, 0, stream>>>(partial, out, P / 4, n_mol);
}